// GATv2Conv_52218212385024
// MI455X (gfx1250) — compile-verified
//
#include <hip/hip_runtime.h>
#include <hip/hip_bf16.h>

#define IN_CH    128
#define HEADS    4
#define HEAD_DIM 16
#define OUT_CH   64        // HEADS * HEAD_DIM
#define NEG_SLOPE 0.2f
#define EPS_F     1e-8f

typedef __attribute__((ext_vector_type(2))) float v2f;
typedef __attribute__((ext_vector_type(8))) float v8f;

// ---------------------------------------------------------------------------
// K0: zero the accumulation buffers (out, att_sum) — must run every launch
// ---------------------------------------------------------------------------
__global__ void gat_zero_kernel(float* __restrict__ out,
                                float* __restrict__ att_sum,
                                int n_nodes) {
    int tid = blockIdx.x * blockDim.x + threadIdx.x;
    if (tid < n_nodes * OUT_CH) out[tid] = 0.0f;
    if (tid < n_nodes * HEADS)  att_sum[tid] = 0.0f;
}

// ---------------------------------------------------------------------------
// K1: h = x @ W^T via V_WMMA_F32_16X16X4_F32.
// One wave per 16x16 output tile; block = 8 waves = 32 nodes x all 64 cols.
// A frag (16x4 f32): lane<16 -> M=lane, K=k+{0,1}; lane>=16 -> M=lane-16, K=k+{2,3}
// B frag (4x16 f32): lane<16 -> N=lane, K=k+{0,1}; lane>=16 -> N=lane-16, K=k+{2,3}
// D (16x16 f32): VGPR r -> M = r + 8*(lane>=16), N = lane&15
// ---------------------------------------------------------------------------
__global__ __launch_bounds__(256) void gat_gemm_wmma_kernel(
    const float* __restrict__ x, const float* __restrict__ W,
    float* __restrict__ h, int n_nodes) {

    const int wave  = threadIdx.x >> 5;
    const int lane  = threadIdx.x & 31;
    const int mtile = (blockIdx.x << 1) + (wave >> 2);
    const int ntile = wave & 3;
    const int mbase = mtile << 4;
    const int nbase = ntile << 4;

    int row_a = mbase + (lane & 15);
    if (row_a >= n_nodes) row_a = n_nodes - 1;       // clamp (no-op for 100000)
    const int koff  = (lane >> 4) << 1;              // 0 or 2
    const int col_b = nbase + (lane & 15);

    const float* __restrict__ xrow = x + (size_t)row_a * IN_CH + koff;
    const float* __restrict__ wrow = W + (size_t)col_b * IN_CH + koff;

    v8f acc = {};
    #pragma unroll
    for (int k = 0; k < IN_CH; k += 4) {
        v2f a; a[0] = xrow[k]; a[1] = xrow[k + 1];
        v2f b; b[0] = wrow[k]; b[1] = wrow[k + 1];
        acc = __builtin_amdgcn_wmma_f32_16x16x4_f32(
            /*neg_a=*/false, a, /*neg_b=*/false, b,
            /*c_mod=*/(short)0, acc, /*reuse_a=*/false, /*reuse_b=*/false);
    }

    const int mrow = mbase + ((lane >> 4) << 3);
    const int ncol = nbase + (lane & 15);
    #pragma unroll
    for (int r = 0; r < 8; ++r) {
        const int m = mrow + r;
        if (m < n_nodes) h[(size_t)m * OUT_CH + ncol] = acc[r];
    }
}

// ---------------------------------------------------------------------------
// K2: per-node attention scalars
//   alpha[n][hd] = ( <a_src[hd], h[n,hd,:]> , <a_dst[hd], h[n,hd,:]> )
// ---------------------------------------------------------------------------
__global__ __launch_bounds__(128) void gat_alpha_kernel(
    const float* __restrict__ h, const float* __restrict__ a,
    float2* __restrict__ alpha, int n_nodes) {

    __shared__ float sa[HEADS * 2 * HEAD_DIM];       // 128 floats
    if (threadIdx.x < HEADS * 2 * HEAD_DIM) sa[threadIdx.x] = a[threadIdx.x];
    __syncthreads();

    const int n = blockIdx.x * blockDim.x + threadIdx.x;
    if (n >= n_nodes) return;

    const float4* __restrict__ hrow = (const float4*)(h + (size_t)n * OUT_CH);
    #pragma unroll
    for (int hd = 0; hd < HEADS; ++hd) {
        float s = 0.0f, d = 0.0f;
        #pragma unroll
        for (int q = 0; q < 4; ++q) {
            const float4 v = hrow[hd * 4 + q];
            const float* as = &sa[hd * 2 * HEAD_DIM + q * 4];
            const float* ad = &sa[hd * 2 * HEAD_DIM + HEAD_DIM + q * 4];
            s += as[0] * v.x + as[1] * v.y + as[2] * v.z + as[3] * v.w;
            d += ad[0] * v.x + ad[1] * v.y + ad[2] * v.z + ad[3] * v.w;
        }
        alpha[(size_t)n * HEADS + hd] = make_float2(s, d);
    }
}

// ---------------------------------------------------------------------------
// K3: single edge pass. Thread = (edge, head); lanes 0..3 share an edge so the
// 256B h_src row gather is coalesced. Accumulate UNNORMALIZED numerator + att
// sum; normalization happens once per node in K4 (same denominator per dst).
// ---------------------------------------------------------------------------
__global__ __launch_bounds__(256) void gat_edge_kernel(
    const int* __restrict__ src_idx, const int* __restrict__ dst_idx,
    const float* __restrict__ ew, const float2* __restrict__ alpha,
    const float* __restrict__ h, float* __restrict__ att_sum,
    float* __restrict__ out, int n_edges) {

    const int tid = blockIdx.x * blockDim.x + threadIdx.x;
    const int e  = tid >> 2;
    const int hd = tid & 3;
    if (e >= n_edges) return;

    const int s = src_idx[e];
    const int d = dst_idx[e];
    const float w = ew[e];

    const float2 as = alpha[(size_t)s * HEADS + hd];
    const float2 ad = alpha[(size_t)d * HEADS + hd];
    float logit = as.x + ad.y;                               // separable GAT logit
    logit = (logit > 0.0f) ? logit : NEG_SLOPE * logit;      // leaky_relu
    const float att = __expf(logit * w);

    atomicAdd(&att_sum[(size_t)d * HEADS + hd], att);

    const float4* __restrict__ hsrc =
        (const float4*)(h + (size_t)s * OUT_CH + hd * HEAD_DIM);
    float* __restrict__ o = out + (size_t)d * OUT_CH + hd * HEAD_DIM;
    #pragma unroll
    for (int q = 0; q < 4; ++q) {
        const float4 v = hsrc[q];
        atomicAdd(&o[q * 4 + 0], att * v.x);
        atomicAdd(&o[q * 4 + 1], att * v.y);
        atomicAdd(&o[q * 4 + 2], att * v.z);
        atomicAdd(&o[q * 4 + 3], att * v.w);
    }
}

// ---------------------------------------------------------------------------
// K4: out[n][hd*16+d] /= (att_sum[n][hd] + eps)
// ---------------------------------------------------------------------------
__global__ __launch_bounds__(256) void gat_norm_kernel(
    float* __restrict__ out, const float* __restrict__ att_sum, int n_nodes) {
    const int tid = blockIdx.x * blockDim.x + threadIdx.x;
    const int n = tid >> 6;
    const int c = tid & 63;
    if (n >= n_nodes) return;
    const int hd = c >> 4;
    const float denom = att_sum[(size_t)n * HEADS + hd] + EPS_F;
    out[(size_t)n * OUT_CH + c] = out[(size_t)n * OUT_CH + c] / denom;
}

// ---------------------------------------------------------------------------
extern "C" void kernel_launch(void* const* d_in, const int* in_sizes, int n_in,
                              void* d_out, int out_size, void* d_ws, size_t ws_size,
                              hipStream_t stream) {
    const float* x  = (const float*)d_in[0];   // (N, 128)
    const int*   ei = (const int*)  d_in[1];   // (2, E)
    const float* ew = (const float*)d_in[2];   // (E,)
    const float* W  = (const float*)d_in[3];   // (64, 128)
    const float* a  = (const float*)d_in[4];   // (1, 4, 32)

    const int n_nodes = in_sizes[0] / IN_CH;
    const int n_edges = in_sizes[2];
    float* out = (float*)d_out;

    // workspace layout
    float*  h       = (float*)d_ws;                              // N*64 f32
    float2* alpha   = (float2*)(h + (size_t)n_nodes * OUT_CH);   // N*4  float2
    float*  att_sum = (float*)(alpha + (size_t)n_nodes * HEADS); // N*4  f32

    // K0: zero accumulators (every launch — deterministic under graph replay)
    {
        const int total = n_nodes * OUT_CH;
        gat_zero_kernel<<<(total + 255) / 256, 256, 0, stream>>>(out, att_sum, n_nodes);
    }
    // K1: WMMA GEMM  h = x @ W^T
    {
        const int blocks = (n_nodes + 31) / 32;   // 32 nodes per block
        gat_gemm_wmma_kernel<<<blocks, 256, 0, stream>>>(x, W, h, n_nodes);
    }
    // K2: per-node attention scalars
    gat_alpha_kernel<<<(n_nodes + 127) / 128, 128, 0, stream>>>(h, a, alpha, n_nodes);
    // K3: edge pass (atomic scatter of numerator + att_sum)
    {
        const long long total = (long long)n_edges * HEADS;
        gat_edge_kernel<<<(unsigned)((total + 255) / 256), 256, 0, stream>>>(
            ei, ei + n_edges, ew, alpha, h, att_sum, out, n_edges);
    }
    // K4: per-node normalization
    {
        const long long total = (long long)n_nodes * OUT_CH;
        gat_norm_kernel<<<(unsigned)((total + 255) / 256), 256, 0, stream>>>(
            out, att_sum, n_nodes);
    }
}